// VisitAttention_26139170963536
// MI455X (gfx1250) — compile-verified
//
#include <hip/hip_runtime.h>
#include <math.h>

typedef __attribute__((ext_vector_type(2))) float v2f;
typedef __attribute__((ext_vector_type(8))) float v8f;

#define B_ 16
#define S_ 2048
#define H_ 256
#define NCHUNK 8
#define CHUNK (S_ / NCHUNK)   /* 256 rows per chunk */
#define EPSF 1e-9f

// ---------------------------------------------------------------------------
// K1: e[b,s] = expf(dot(x[b,s,:], W) + bias). One wave32 per row; each lane
// handles 8 of the 256 H elements, then a 5-step shfl_xor reduction.
// ---------------------------------------------------------------------------
__global__ void k_scores(const float* __restrict__ x, const float* __restrict__ W,
                         const float* __restrict__ bias, float* __restrict__ e) {
    int gt   = blockIdx.x * blockDim.x + threadIdx.x;
    int row  = gt >> 5;
    int lane = gt & 31;
    const float4* xr = reinterpret_cast<const float4*>(x + (size_t)row * H_ + lane * 8);
    const float4* wr = reinterpret_cast<const float4*>(W + lane * 8);
    float4 a0 = xr[0], a1 = xr[1];
    float4 w0 = wr[0], w1 = wr[1];
    float d = a0.x*w0.x + a0.y*w0.y + a0.z*w0.z + a0.w*w0.w
            + a1.x*w1.x + a1.y*w1.y + a1.z*w1.z + a1.w*w1.w;
    #pragma unroll
    for (int off = 16; off > 0; off >>= 1) d += __shfl_xor(d, off, 32);
    if (lane == 0) e[row] = expf(d + bias[0]);
}

// ---------------------------------------------------------------------------
// K2: csum[b,i] = inclusive cumsum of e over S, + EPS. One block per batch,
// 256 threads x 8 elements, Hillis-Steele block scan in LDS.
// ---------------------------------------------------------------------------
__global__ void k_cumsum(const float* __restrict__ e, float* __restrict__ csum) {
    __shared__ float sd[256];
    int b = blockIdx.x, t = threadIdx.x;
    const float* eb = e + (size_t)b * S_ + t * 8;
    float4 p0 = reinterpret_cast<const float4*>(eb)[0];
    float4 p1 = reinterpret_cast<const float4*>(eb)[1];
    float v[8] = {p0.x, p0.y, p0.z, p0.w, p1.x, p1.y, p1.z, p1.w};
    #pragma unroll
    for (int k = 1; k < 8; ++k) v[k] += v[k-1];
    float tot = v[7];
    sd[t] = tot;
    __syncthreads();
    float run = tot;
    for (int off = 1; off < 256; off <<= 1) {
        float y = (t >= off) ? sd[t - off] : 0.0f;
        __syncthreads();
        run += y;
        sd[t] = run;
        __syncthreads();
    }
    float excl = run - tot;
    float* cb = csum + (size_t)b * S_ + t * 8;
    #pragma unroll
    for (int k = 0; k < 8; ++k) cb[k] = excl + v[k] + EPSF;
}

// ---------------------------------------------------------------------------
// K3a: per-chunk vector sums  chunkSum[b,c,h] = sum_{j in chunk c} e[j]*x[j,h]
// ---------------------------------------------------------------------------
__global__ void k_chunksum(const float* __restrict__ x, const float* __restrict__ e,
                           float* __restrict__ chunkSum) {
    int b = blockIdx.x / NCHUNK;
    int c = blockIdx.x % NCHUNK;
    int h = threadIdx.x;
    const float* eb = e + (size_t)b * S_;
    const float* xb = x + ((size_t)b * S_) * H_;
    float acc = 0.0f;
    int row0 = c * CHUNK;
    for (int r = 0; r < CHUNK; ++r) {
        int row = row0 + r;
        acc += eb[row] * xb[(size_t)row * H_ + h];
    }
    chunkSum[((size_t)b * NCHUNK + c) * H_ + h] = acc;
}

// ---------------------------------------------------------------------------
// K3b: exclusive scan of chunk sums along chunks -> per-chunk carry-in
// ---------------------------------------------------------------------------
__global__ void k_carry(const float* __restrict__ chunkSum, float* __restrict__ carry) {
    int b = blockIdx.x, h = threadIdx.x;
    float run = 0.0f;
    for (int c = 0; c < NCHUNK; ++c) {
        size_t idx = ((size_t)b * NCHUNK + c) * H_ + h;
        carry[idx] = run;
        run += chunkSum[idx];
    }
}

// ---------------------------------------------------------------------------
// K3c: WMMA tile scan. One wave handles (batch, chunk, 16-wide h-tile) and
// walks 16 row-tiles. Per tile:  P = L16 @ Y + broadcast(carry), via
// 4x V_WMMA_F32_16X16X4_F32; out = P / csum; carry <- row 15 of P.
// A 16x4 f32 layout: lanes 0-15 {K0,K1}, lanes 16-31 {K2,K3}, M = lane&15.
// B 4x16 f32 layout: same K split, N = lane&15.
// C/D 16x16 f32:     M = vgpr + 8*(lane>=16), N = lane&15.
// ---------------------------------------------------------------------------
__global__ void k_wout(const float* __restrict__ x, const float* __restrict__ e,
                       const float* __restrict__ csum, const float* __restrict__ carryWS,
                       float* __restrict__ out) {
    int wid  = (blockIdx.x * blockDim.x + threadIdx.x) >> 5;
    int lane = threadIdx.x & 31;
    int ht    = wid & 15;          // H_/16 = 16 tiles
    int chunk = (wid >> 4) & 7;    // NCHUNK = 8
    int b     = wid >> 7;          // B_ = 16
    int col = lane & 15;
    int hi  = lane >> 4;
    int hcol = ht * 16 + col;

    const float* eb = e    + (size_t)b * S_;
    const float* cb = csum + (size_t)b * S_;
    const float* xb = x    + ((size_t)b * S_) * H_;
    float*       ob = out  + ((size_t)b * S_) * H_;

    float carry = carryWS[((size_t)b * NCHUNK + chunk) * H_ + hcol];

    // Constant A-matrix K-slices of the 16x16 lower-triangular ones matrix.
    v2f a[4];
    #pragma unroll
    for (int k = 0; k < 4; ++k) {
        int kg = 4 * k + 2 * hi;
        a[k].x = (kg     <= col) ? 1.0f : 0.0f;
        a[k].y = (kg + 1 <= col) ? 1.0f : 0.0f;
    }

    int row0 = chunk * CHUNK;
    for (int t = 0; t < CHUNK / 16; ++t) {
        int r0 = row0 + t * 16;
        v8f c;
        #pragma unroll
        for (int v = 0; v < 8; ++v) c[v] = carry;   // C[i,h] = carry[h] for all i
        #pragma unroll
        for (int k = 0; k < 4; ++k) {
            int j0 = r0 + 4 * k + 2 * hi;
            v2f bm;
            bm.x = eb[j0]     * xb[(size_t)j0       * H_ + hcol];
            bm.y = eb[j0 + 1] * xb[(size_t)(j0 + 1) * H_ + hcol];
            c = __builtin_amdgcn_wmma_f32_16x16x4_f32(false, a[k], false, bm,
                                                      (short)0, c, false, false);
        }
        #pragma unroll
        for (int v = 0; v < 8; ++v) {
            int i = r0 + v + 8 * hi;
            ob[(size_t)i * H_ + hcol] = c[v] / cb[i];
        }
        // carry <- inclusive prefix through row 15 = D[M=15][N], in vgpr7 of hi half
        carry = __shfl(c[7], 16 + col, 32);
    }
}

// ---------------------------------------------------------------------------
// K4: attention_weight[b,i,j] = (j<=i) ? e[b,j]/csum[b,i] : 0.
// 268 MB of stores -> the bandwidth-dominant kernel. One block per output
// row, float4 streaming stores (zeros included: d_out is poisoned).
// ---------------------------------------------------------------------------
__global__ void k_attn(const float* __restrict__ e, const float* __restrict__ csum,
                       float* __restrict__ attn) {
    int bi = blockIdx.x;        // b*S + i
    int b  = bi >> 11;          // S_ = 2048
    int i  = bi & (S_ - 1);
    float rinv = 1.0f / csum[(size_t)b * S_ + i];
    const float4* e4 = reinterpret_cast<const float4*>(e + (size_t)b * S_);
    float4*       o4 = reinterpret_cast<float4*>(attn + (size_t)bi * S_);
    #pragma unroll
    for (int it = 0; it < 2; ++it) {
        int q  = threadIdx.x + it * 256;  // float4 index 0..511
        int j0 = q * 4;
        float4 r;
        if (j0 + 3 <= i) {
            float4 ev = e4[q];
            r.x = ev.x * rinv; r.y = ev.y * rinv;
            r.z = ev.z * rinv; r.w = ev.w * rinv;
        } else if (j0 > i) {
            r.x = 0.0f; r.y = 0.0f; r.z = 0.0f; r.w = 0.0f;
        } else {
            float4 ev = e4[q];
            r.x = (j0     <= i) ? ev.x * rinv : 0.0f;
            r.y = (j0 + 1 <= i) ? ev.y * rinv : 0.0f;
            r.z = (j0 + 2 <= i) ? ev.z * rinv : 0.0f;
            r.w = (j0 + 3 <= i) ? ev.w * rinv : 0.0f;
        }
        o4[q] = r;
    }
}

// ---------------------------------------------------------------------------
extern "C" void kernel_launch(void* const* d_in, const int* in_sizes, int n_in,
                              void* d_out, int out_size, void* d_ws, size_t ws_size,
                              hipStream_t stream) {
    const float* x    = (const float*)d_in[0];   // [B,S,H]
    const float* W    = (const float*)d_in[1];   // [H,1]
    const float* bias = (const float*)d_in[2];   // [1]

    float* wout = (float*)d_out;                           // [B,S,H]
    float* attn = (float*)d_out + (size_t)B_ * S_ * H_;    // [B,S,S]

    float* e        = (float*)d_ws;                        // B*S
    float* csum     = e + (size_t)B_ * S_;                 // B*S
    float* chunkSum = csum + (size_t)B_ * S_;              // B*NCHUNK*H
    float* carry    = chunkSum + (size_t)B_ * NCHUNK * H_; // B*NCHUNK*H

    k_scores  <<<(B_ * S_) / 8,      256, 0, stream>>>(x, W, bias, e);
    k_cumsum  <<<B_,                 256, 0, stream>>>(e, csum);
    k_chunksum<<<B_ * NCHUNK,        256, 0, stream>>>(x, e, chunkSum);
    k_carry   <<<B_,                 256, 0, stream>>>(chunkSum, carry);
    // 16 batches * 8 chunks * 16 h-tiles = 2048 waves = 256 blocks of 8 waves
    k_wout    <<<(B_ * NCHUNK * (H_ / 16)) * 32 / 256, 256, 0, stream>>>(
                  x, e, csum, carry, wout);
    k_attn    <<<B_ * S_,            256, 0, stream>>>(e, csum, attn);
}